// Multiplexer_36258113913305
// MI455X (gfx1250) — compile-verified
//
#include <hip/hip_runtime.h>

// Per-row contiguous 64-float block gather:
//   out[b, 0:64] = full_input[b, idx[b]*64 : idx[b]*64 + 64]
// BATCH = 262144, row length = 1024 floats, block = 64 floats.
//
// Pure bandwidth problem (0 FLOPs, ~129 MB of traffic -> ~5.5us at 23.3 TB/s).
// Strategy: 1 float4 per thread, 16 threads per row -> coalesced 128-bit
// loads/stores; non-temporal hints since input (1 GB) and output (64 MB) are
// touched exactly once and should not displace the 192 MB L2's useful state.

typedef __attribute__((ext_vector_type(4))) float v4f;

#define OUTPUT_DIM 64
#define ROW_F4 256   // 1024 floats per input row = 256 float4
#define BLK_F4 16    // 64 floats per selected block = 16 float4

__global__ __launch_bounds__(256) void mux_gather_kernel(
    const float* __restrict__ full_input,
    const int* __restrict__ indices,
    float* __restrict__ out,
    unsigned n_rows) {
  // global float4 slot id: 16 slots per row
  unsigned g = blockIdx.x * 256u + threadIdx.x;
  unsigned row = g >> 4;      // [0, BATCH)
  unsigned part = g & 15u;    // [0, 16)  float4 within the 64-float block
  if (row >= n_rows) return;

  int idx = indices[row];     // [0, 16)

  const v4f* __restrict__ src =
      (const v4f*)full_input + (size_t)row * ROW_F4 + (unsigned)idx * BLK_F4 + part;
  v4f* __restrict__ dst = (v4f*)out + (size_t)row * BLK_F4 + part;

  // Read-once / write-once streaming: non-temporal on both sides
  // (expect global_load_b128 th:TH_LOAD_NT / global_store_b128 th:TH_STORE_NT).
  v4f v = __builtin_nontemporal_load(src);
  __builtin_nontemporal_store(v, dst);
}

extern "C" void kernel_launch(void* const* d_in, const int* in_sizes, int n_in,
                              void* d_out, int out_size, void* d_ws, size_t ws_size,
                              hipStream_t stream) {
  const float* full_input = (const float*)d_in[0];  // [B, 1024] fp32
  const int* indices      = (const int*)d_in[1];    // [B, 1] int32
  float* out              = (float*)d_out;          // [B, 64] fp32

  // rows = out_size / OUTPUT_DIM
  unsigned n_rows = (unsigned)(out_size / OUTPUT_DIM);
  unsigned total_threads = n_rows * BLK_F4;         // one float4 per thread
  unsigned blocks = (total_threads + 255u) / 256u;

  mux_gather_kernel<<<blocks, 256, 0, stream>>>(full_input, indices, out, n_rows);
}